// SigLoss_naive_42984032698500
// MI455X (gfx1250) — compile-verified
//
#include <hip/hip_runtime.h>

typedef __attribute__((ext_vector_type(2))) float v2f;
typedef __attribute__((ext_vector_type(8))) float v8f;

#define A_BATCH 128
#define M_PTS   256
#define D_DIM   64
#define MI      255          // inc rows/cols = M-1
#define DX_STR  65           // padded LDS row stride (floats) for dX/dY
#define T_STR   260          // inc tile col stride (floats)
#define NBLK    16           // 16 row-blocks of 16 rows cover 256 (row 255 is zero pad)

// One workgroup per (pair, batch): 3*128 = 384 WGs, 256 threads (8 wave32).
// Fused: WMMA f32 computes a 16x256 inc tile into LDS, wave0 runs the
// prefix-scan form of the signature-kernel PDE over those 16 rows.
__global__ __launch_bounds__(256) void sig_pde_kernel(const float* __restrict__ X,
                                                      const float* __restrict__ Y,
                                                      float* __restrict__ partial) {
  const int wg = blockIdx.x;          // 0..383
  const int p  = wg >> 7;             // 0: xx, 1: yy, 2: xy
  const int a  = wg & 127;
  const float* P = (p == 1) ? Y : X;  // dX side
  const float* Q = (p == 0) ? X : Y;  // dY side
  const float* Pa = P + (size_t)a * M_PTS * D_DIM;
  const float* Qa = Q + (size_t)a * M_PTS * D_DIM;

  extern __shared__ float smem[];
  float* sdx  = smem;                        // 256 x 65
  float* sdy  = smem + 256 * DX_STR;         // 256 x 65
  float* tile = smem + 2 * 256 * DX_STR;     // 16 x 260 (col 0 = boundary zeros)

  const int lane = threadIdx.x & 31;
  const int wv   = threadIdx.x >> 5;

  // ---- Load path increments into LDS (padded stride kills bank conflicts) ----
  for (int idx = threadIdx.x; idx < MI * D_DIM; idx += blockDim.x) {
    const int i = idx >> 6;
    const int k = idx & 63;
    sdx[i * DX_STR + k] = Pa[(i + 1) * D_DIM + k] - Pa[i * D_DIM + k];
    sdy[i * DX_STR + k] = Qa[(i + 1) * D_DIM + k] - Qa[i * D_DIM + k];
  }
  for (int k = threadIdx.x; k < D_DIM; k += blockDim.x) {   // zero pad row 255
    sdx[255 * DX_STR + k] = 0.0f;
    sdy[255 * DX_STR + k] = 0.0f;
  }
  if (threadIdx.x < NBLK) tile[threadIdx.x * T_STR] = 0.0f; // boundary column
  __syncthreads();

  // ---- Wave0 scan state: K row in registers, 8 columns per lane ----
  float kp[8];
#pragma unroll
  for (int e = 0; e < 8; ++e) kp[e] = 1.0f;   // K[0][:] = 1

  // WMMA fragment coordinates (16x16x4 f32):
  //  A 16x4 : lanes 0-15 -> M=lane, K = {0,1}; lanes 16-31 -> M=lane-16, K = {2,3}
  //  B 4x16 : lanes 0-15 -> N=lane, K = {0,1}; lanes 16-31 -> N=lane-16, K = {2,3}
  //  C 16x16: lane n (0-15)+v -> (M=v, N=n); lanes 16-31 -> (M=v+8, N=n-16)
  const int mn   = lane & 15;
  const int koff = (lane >> 4) * 2;

  for (int rb = 0; rb < NBLK; ++rb) {
    const int arow = rb * 16 + mn;

    // Hoist all A fragments for this row-block (shared by both N-tiles):
    // 16 batched ds loads, one wait.
    v2f afrag[16];
#pragma unroll
    for (int kk = 0; kk < 16; ++kk) {
      afrag[kk].x = sdx[arow * DX_STR + kk * 4 + koff];
      afrag[kk].y = sdx[arow * DX_STR + kk * 4 + koff + 1];
    }

#pragma unroll
    for (int tt = 0; tt < 2; ++tt) {
      const int t    = wv * 2 + tt;          // N-tile 0..15
      const int bcol = t * 16 + mn;

      // Hoist all B fragments for this tile too: batched loads + single
      // wait, then 16 back-to-back v_wmma accumulates (no per-op waits).
      v2f bfrag[16];
#pragma unroll
      for (int kk = 0; kk < 16; ++kk) {
        bfrag[kk].x = sdy[bcol * DX_STR + kk * 4 + koff];
        bfrag[kk].y = sdy[bcol * DX_STR + kk * 4 + koff + 1];
      }

      v8f c = {};
#pragma unroll
      for (int kk = 0; kk < 16; ++kk) {
        c = __builtin_amdgcn_wmma_f32_16x16x4_f32(
            /*neg_a=*/false, afrag[kk], /*neg_b=*/false, bfrag[kk],
            /*c_mod=*/(short)0, c, /*reuse_a=*/false, /*reuse_b=*/false);
      }

      // Store C to LDS tile shifted +1 column: tile[r][j] = inc[gi][j-1]
      const int rbase = (lane >> 4) * 8;
#pragma unroll
      for (int v = 0; v < 8; ++v) {
        tile[(rbase + v) * T_STR + t * 16 + mn + 1] = c[v];
      }
    }
    __syncthreads();

    // ---- Wave0: PDE rows as prefix scans ----
    if (wv == 0) {
      const int rmax = (rb == NBLK - 1) ? 15 : 16;   // inc row 255 doesn't exist
      for (int r = 0; r < rmax; ++r) {
        // K_prev[8*lane - 1] from left neighbor
        float km1 = __shfl_up(kp[7], 1);
        float loc[8];
        float run = 0.0f;
#pragma unroll
        for (int e = 0; e < 8; ++e) {
          const float incv = tile[r * T_STR + lane * 8 + e];   // inc[gi][j-1]
          const float km   = (e == 0) ? km1 : kp[e - 1];
          float cj = kp[e] + km * (incv - 1.0f);
          if (lane == 0 && e == 0) cj = 0.0f;                  // j=0 boundary
          run += cj;
          loc[e] = run;
        }
        // wave32 inclusive scan of per-lane sums
        float tot = run;
#pragma unroll
        for (int off = 1; off < 32; off <<= 1) {
          float v = __shfl_up(tot, off);
          if (lane >= off) tot += v;
        }
        const float excl = tot - run;
#pragma unroll
        for (int e = 0; e < 8; ++e) kp[e] = 1.0f + excl + loc[e];
      }
    }
    __syncthreads();
  }

  if (wv == 0 && lane == 31) partial[wg] = kp[7];   // K[255][255]
}

// Deterministic fixed-order reduction: mean(kxx) + mean(kyy) - 2*mean(kxy)
__global__ void sig_reduce_kernel(const float* __restrict__ partial,
                                  float* __restrict__ out) {
  if (threadIdx.x == 0 && blockIdx.x == 0) {
    float sxx = 0.0f, syy = 0.0f, sxy = 0.0f;
    for (int a = 0; a < A_BATCH; ++a) {
      sxx += partial[a];
      syy += partial[A_BATCH + a];
      sxy += partial[2 * A_BATCH + a];
    }
    const float inv = 1.0f / (float)A_BATCH;
    out[0] = sxx * inv + syy * inv - 2.0f * sxy * inv;
  }
}

extern "C" void kernel_launch(void* const* d_in, const int* in_sizes, int n_in,
                              void* d_out, int out_size, void* d_ws, size_t ws_size,
                              hipStream_t stream) {
  const float* x = (const float*)d_in[0];
  const float* y = (const float*)d_in[1];
  float* ws  = (float*)d_ws;    // 384 floats of per-problem results
  float* out = (float*)d_out;

  const size_t sh = (size_t)(2 * 256 * DX_STR + NBLK * T_STR) * sizeof(float);
  hipFuncSetAttribute((const void*)sig_pde_kernel,
                      hipFuncAttributeMaxDynamicSharedMemorySize, (int)sh);

  sig_pde_kernel<<<dim3(3 * A_BATCH), dim3(256), sh, stream>>>(x, y, ws);
  sig_reduce_kernel<<<dim3(1), dim3(32), 0, stream>>>(ws, out);
}